// RG_LRU_7662221656705
// MI455X (gfx1250) — compile-verified
//
#include <hip/hip_runtime.h>
#include <math.h>

typedef __attribute__((ext_vector_type(2))) float v2f;
typedef __attribute__((ext_vector_type(4))) float v4f;
typedef __attribute__((ext_vector_type(8))) float v8f;

#define DD 2048
#define BB 16
#define KSPLIT 8
#define CHUNK (DD / KSPLIT)      // 256 k per wave
#define STEPS (CHUNK / 8)        // 32 steps of 8 physical k
#define PDEPTH 4                 // register prefetch depth (stages of 8 k)

// ---------------------------------------------------------------------------
// Kernel 1: partial GEMMs z_a = xt@Wa.T, z_b = xt@Wx.T via f32 WMMA.
// 128 N-tiles x 8 K-splits = 1024 waves. Each wave: 32 steps of 8 k,
// software-pipelined with a 4-stage rotating register buffer, b128 loads.
// Virtual-K remap: per 8-k group, lane loads float4 at kg + 4*half;
// WMMA step0 uses f4[0],f4[1]; step1 uses f4[2],f4[3] (A and B share the
// same physical-k mapping, so the K-reduction is unchanged).
// ---------------------------------------------------------------------------
__global__ __launch_bounds__(256) void rg_lru_gemm_kernel(
    const float* __restrict__ xt,   // [16, 2048]
    const float* __restrict__ Wa,   // [2048, 2048] row-major
    const float* __restrict__ Wx,   // [2048, 2048]
    float* __restrict__ part_a,     // [KSPLIT, 16, 2048]
    float* __restrict__ part_b)     // [KSPLIT, 16, 2048]
{
  const int lane  = threadIdx.x & 31;
  const int wgl   = blockIdx.x * (blockDim.x >> 5) + (threadIdx.x >> 5); // 0..1023
  const int tile  = wgl >> 3;          // 0..127 -> N-tile
  const int split = wgl & (KSPLIT - 1);
  const int n0    = tile << 4;
  const int k0    = split * CHUNK;
  const int half  = lane >> 4;
  const int l     = lane & 15;

  const float* xrow  = xt + l * DD + k0 + 4 * half;                  // A: m = l
  const float* warow = Wa + (size_t)(n0 + l) * DD + k0 + 4 * half;   // B[k][n] = Wa[n][k]
  const float* wxrow = Wx + (size_t)(n0 + l) * DD + k0 + 4 * half;

  v4f abuf[PDEPTH], wabuf[PDEPTH], wxbuf[PDEPTH];
#pragma unroll
  for (int p = 0; p < PDEPTH; ++p) {
    const int kg = p * 8;
    abuf[p]  = *(const v4f*)(xrow  + kg);
    wabuf[p] = *(const v4f*)(warow + kg);
    wxbuf[p] = *(const v4f*)(wxrow + kg);
  }

  v8f acc_a = {};
  v8f acc_b = {};

#pragma unroll 1
  for (int it = 0; it < STEPS - PDEPTH; it += PDEPTH) {
#pragma unroll
    for (int p = 0; p < PDEPTH; ++p) {
      v4f ca  = abuf[p];
      v4f cwa = wabuf[p];
      v4f cwx = wxbuf[p];
      const int kl = (it + p + PDEPTH) * 8;   // prefetch PDEPTH stages ahead
      abuf[p]  = *(const v4f*)(xrow  + kl);
      wabuf[p] = *(const v4f*)(warow + kl);
      wxbuf[p] = *(const v4f*)(wxrow + kl);

      v2f a01  = __builtin_shufflevector(ca,  ca,  0, 1);
      v2f a23  = __builtin_shufflevector(ca,  ca,  2, 3);
      v2f wa01 = __builtin_shufflevector(cwa, cwa, 0, 1);
      v2f wa23 = __builtin_shufflevector(cwa, cwa, 2, 3);
      v2f wx01 = __builtin_shufflevector(cwx, cwx, 0, 1);
      v2f wx23 = __builtin_shufflevector(cwx, cwx, 2, 3);
      acc_a = __builtin_amdgcn_wmma_f32_16x16x4_f32(false, a01, false, wa01, (short)0, acc_a, false, false);
      acc_b = __builtin_amdgcn_wmma_f32_16x16x4_f32(false, a01, false, wx01, (short)0, acc_b, false, false);
      acc_a = __builtin_amdgcn_wmma_f32_16x16x4_f32(false, a23, false, wa23, (short)0, acc_a, false, false);
      acc_b = __builtin_amdgcn_wmma_f32_16x16x4_f32(false, a23, false, wx23, (short)0, acc_b, false, false);
    }
  }
  // drain the pipeline
#pragma unroll
  for (int p = 0; p < PDEPTH; ++p) {
    v4f ca  = abuf[p];
    v4f cwa = wabuf[p];
    v4f cwx = wxbuf[p];
    v2f a01  = __builtin_shufflevector(ca,  ca,  0, 1);
    v2f a23  = __builtin_shufflevector(ca,  ca,  2, 3);
    v2f wa01 = __builtin_shufflevector(cwa, cwa, 0, 1);
    v2f wa23 = __builtin_shufflevector(cwa, cwa, 2, 3);
    v2f wx01 = __builtin_shufflevector(cwx, cwx, 0, 1);
    v2f wx23 = __builtin_shufflevector(cwx, cwx, 2, 3);
    acc_a = __builtin_amdgcn_wmma_f32_16x16x4_f32(false, a01, false, wa01, (short)0, acc_a, false, false);
    acc_b = __builtin_amdgcn_wmma_f32_16x16x4_f32(false, a01, false, wx01, (short)0, acc_b, false, false);
    acc_a = __builtin_amdgcn_wmma_f32_16x16x4_f32(false, a23, false, wa23, (short)0, acc_a, false, false);
    acc_b = __builtin_amdgcn_wmma_f32_16x16x4_f32(false, a23, false, wx23, (short)0, acc_b, false, false);
  }

  // C/D layout: VGPR v, lane half -> m = v + 8*half, col = n0 + (lane&15)
  const int col = n0 + l;
#pragma unroll
  for (int v = 0; v < 8; ++v) {
    const int m = v + 8 * half;
    part_a[((size_t)split * BB + m) * DD + col] = acc_a[v];
    part_b[((size_t)split * BB + m) * DD + col] = acc_b[v];
  }
}

// ---------------------------------------------------------------------------
// Kernel 2: reduce K-split partials + gate math -> a[b,n], u[b,n].
// ---------------------------------------------------------------------------
__global__ __launch_bounds__(256) void rg_lru_gate_kernel(
    const float* __restrict__ part_a,
    const float* __restrict__ part_b,
    const float* __restrict__ xt,
    const float* __restrict__ ba,
    const float* __restrict__ bx,
    const float* __restrict__ Lam,
    float* __restrict__ a_ws,
    float* __restrict__ u_ws)
{
  const int idx = blockIdx.x * blockDim.x + threadIdx.x;  // 0 .. 16*2048-1
  if (idx >= BB * DD) return;
  const int col = idx & (DD - 1);

  float za = ba[col];
  float zb = bx[col];
#pragma unroll
  for (int s = 0; s < KSPLIT; ++s) {
    za += part_a[(size_t)s * BB * DD + idx];
    zb += part_b[(size_t)s * BB * DD + idx];
  }
  float rt = 1.0f / (1.0f + expf(-za));
  float it = 1.0f / (1.0f + expf(-zb));
  float log_a = -log1pf(expf(-Lam[col]));     // -softplus(-Lam)
  float a = expf(log_a * rt * 0.125f);        // /C, C=8
  float u = sqrtf(fmaxf(1.0f - a * a, 0.0f)) * (it * xt[idx]);
  a_ws[idx] = a;
  u_ws[idx] = u;
}

// ---------------------------------------------------------------------------
// Kernel 3: stream out ht[b,i,:] = u[b,:] (+ a[b,i]*h[b,i] on the diagonal).
// 256 MiB of streaming writes -> non-temporal b128 stores (output > 192MB L2);
// u rows (128 KB) remain L2-resident across the 2048x reuse.
// ---------------------------------------------------------------------------
__global__ __launch_bounds__(256) void rg_lru_fill_kernel(
    const float* __restrict__ a_ws,  // [16, 2048]
    const float* __restrict__ u_ws,  // [16, 2048]
    const float* __restrict__ h,     // ht_minus_1 [16, 2048]
    float* __restrict__ out)         // [16, 2048, 2048]
{
  const int row = blockIdx.x;        // 0 .. 16*2048-1
  const int b   = row >> 11;
  const int i   = row & (DD - 1);
  const float diag = a_ws[b * DD + i] * h[b * DD + i];

  const v4f* u4 = (const v4f*)(u_ws + (size_t)b * DD);
  v4f*       o4 = (v4f*)(out + (size_t)row * DD);

  for (int t = threadIdx.x; t < DD / 4; t += blockDim.x) {
    v4f val = u4[t];
    const int j0 = t << 2;
    if (i >= j0 && i < j0 + 4) val[i - j0] += diag;
    __builtin_nontemporal_store(val, o4 + t);
  }
}

extern "C" void kernel_launch(void* const* d_in, const int* in_sizes, int n_in,
                              void* d_out, int out_size, void* d_ws, size_t ws_size,
                              hipStream_t stream) {
  // setup_inputs order: xt, ht_minus_1, Wa, Wx, ba, bx, Lam
  const float* xt  = (const float*)d_in[0];
  const float* h   = (const float*)d_in[1];
  const float* Wa  = (const float*)d_in[2];
  const float* Wx  = (const float*)d_in[3];
  const float* ba  = (const float*)d_in[4];
  const float* bx  = (const float*)d_in[5];
  const float* Lam = (const float*)d_in[6];
  float* out = (float*)d_out;

  const size_t NP = (size_t)KSPLIT * BB * DD;   // 256K floats per partial array
  float* part_a = (float*)d_ws;                 // 1 MB
  float* part_b = part_a + NP;                  // 1 MB
  float* a_ws   = part_b + NP;                  // 128 KB
  float* u_ws   = a_ws + (size_t)BB * DD;       // 128 KB

  // 1024 waves = 128 blocks x 256 threads (8 waves/block).
  rg_lru_gemm_kernel<<<dim3(128), dim3(256), 0, stream>>>(
      xt, Wa, Wx, part_a, part_b);

  // 32768 elements -> 128 blocks x 256 threads.
  rg_lru_gate_kernel<<<dim3(128), dim3(256), 0, stream>>>(
      part_a, part_b, xt, ba, bx, Lam, a_ws, u_ws);

  // One block per output row (b,i): 16*2048 = 32768 blocks.
  rg_lru_fill_kernel<<<dim3(BB * DD), dim3(256), 0, stream>>>(
      a_ws, u_ws, h, out);
}